// GNNIntraAgg_43250320670866
// MI455X (gfx1250) — compile-verified
//
#include <hip/hip_runtime.h>
#include <stdint.h>

// GNN neighbor-mean + relu:  out[b,:] = relu( (1/32) * sum_k features[neigh_ids[b,k], :] )
// B=16384, K=32, N=100000, D=256 (f32).
//
// Roofline: ~134 MFLOP vs ~530 MB moved -> purely memory bound. The 102.4 MB
// feature table is L2-resident (192 MB L2), so the kernel is an L2-bandwidth /
// latency-hiding problem. CDNA5 path: ASYNCcnt-tracked global->LDS DMA gather
// (global_load_async_to_lds_b128) with an 8-deep per-wave ring buffer and a
// consume-then-reissue schedule that keeps 16 async ops (8 KB) in flight per wave.
// WMMA is deliberately not used: the reduction is a batched gather-reduce (b and d
// both free), which the f32 16x16x4 WMMA can only express with 16x-redundant
// columns + cross-lane marshalling — strictly more memory work on a memory-bound
// kernel, and bf16/f16 WMMA would degrade f32 precision.

constexpr int K_NEIGH = 32;    // neighbors per row
constexpr int D_FEAT  = 256;   // floats per feature row (1 KB)
constexpr int WAVES   = 8;     // waves per block (wave32)
constexpr int PIPE    = 8;     // LDS ring-buffer depth per wave (8 KB/wave)
constexpr int LEAD    = 8;     // prefetch distance in rows (16 async ops in flight)

// Issue one 1KB feature row: 2 async ops, each moves 32 lanes x 16B = 512B.
// Per CDNA5 ISA, the instruction offset is added to BOTH the global and LDS address.
__device__ __forceinline__ void async_row(const float* g, uint32_t lds_off) {
  asm volatile(
      "global_load_async_to_lds_b128 %0, %1, off\n\t"
      "global_load_async_to_lds_b128 %0, %1, off offset:512"
      :: "v"(lds_off), "v"(g)
      : "memory");
}

// Fully unrolled pipeline so every s_wait_asynccnt immediate is a compile-time
// constant (correct drained counts in the tail).
template <int K>
__device__ __forceinline__ void pipe_all(const float* __restrict__ feat, int myid,
                                         const float* __restrict__ wbuf,
                                         uint32_t wbase_lane, int lane,
                                         float4& a0, float4& a1) {
  if constexpr (K < K_NEIGH) {
    constexpr int rem  = K_NEIGH - 1 - K;                      // rows after this one
    constexpr int wcnt = 2 * (rem < (LEAD - 1) ? rem : (LEAD - 1));
    // Wait until row K's two async ops have landed in LDS
    // (async-load completion is in-order, so ASYNCcnt <= wcnt implies row K done).
    asm volatile("s_wait_asynccnt %0" :: "i"(wcnt) : "memory");

    // Consume row K from LDS: lane owns float4 slots [lane] and [lane+32].
    const float* lrow = wbuf + (size_t)((K & (PIPE - 1)) * D_FEAT);
    float4 x0 = *(const float4*)(lrow + lane * 4);
    float4 x1 = *(const float4*)(lrow + (D_FEAT / 2) + lane * 4);
    a0.x += x0.x; a0.y += x0.y; a0.z += x0.z; a0.w += x0.w;
    a1.x += x1.x; a1.y += x1.y; a1.z += x1.z; a1.w += x1.w;

    // Re-issue into the ring slot just freed ((K+LEAD) & 7 == K & 7).
    // The "memory" clobber on the asm keeps the LDS reads above ordered before it.
    if constexpr (K + LEAD < K_NEIGH) {
      int sid = __builtin_amdgcn_readlane(myid, K + LEAD);
      const float* g = feat + (size_t)sid * D_FEAT + lane * 4;
      async_row(g, wbase_lane + (uint32_t)(((K + LEAD) & (PIPE - 1)) * (D_FEAT * 4)));
    }

    pipe_all<K + 1>(feat, myid, wbuf, wbase_lane, lane, a0, a1);
  }
}

__global__ __launch_bounds__(WAVES * 32)
void gnn_mean_relu_kernel(const int* __restrict__ neigh_ids,
                          const float* __restrict__ feat,
                          float* __restrict__ out, int Brows) {
  __shared__ float lds[WAVES][PIPE][D_FEAT];

  const int tid  = threadIdx.x;
  const int wave = tid >> 5;
  const int lane = tid & 31;
  const int b    = blockIdx.x * WAVES + wave;
  if (b >= Brows) return;

  const float* wbuf = &lds[wave][0][0];
  // Low 32 bits of the flat shared-aperture address == LDS byte offset.
  const uint32_t wbase_lane =
      (uint32_t)(uintptr_t)wbuf + (uint32_t)(lane * 16);

  // Lane L holds neighbor id L for this row (coalesced 128B load per wave).
  const int myid = neigh_ids[(size_t)b * K_NEIGH + lane];

  // Prologue: fill the whole ring (rows 0..LEAD-1 -> 16 async ops in flight).
#pragma unroll
  for (int k = 0; k < LEAD; ++k) {
    int sid = __builtin_amdgcn_readlane(myid, k);
    const float* g = feat + (size_t)sid * D_FEAT + lane * 4;
    async_row(g, wbase_lane + (uint32_t)(k * (D_FEAT * 4)));
  }

  float4 a0 = {0.f, 0.f, 0.f, 0.f};
  float4 a1 = {0.f, 0.f, 0.f, 0.f};
  pipe_all<0>(feat, myid, wbuf, wbase_lane, lane, a0, a1);

  // mean + relu, then two coalesced 512B stores per wave.
  const float s = 1.0f / (float)K_NEIGH;
  float4 r0, r1;
  r0.x = fmaxf(a0.x * s, 0.f); r0.y = fmaxf(a0.y * s, 0.f);
  r0.z = fmaxf(a0.z * s, 0.f); r0.w = fmaxf(a0.w * s, 0.f);
  r1.x = fmaxf(a1.x * s, 0.f); r1.y = fmaxf(a1.y * s, 0.f);
  r1.z = fmaxf(a1.z * s, 0.f); r1.w = fmaxf(a1.w * s, 0.f);

  float4* o = (float4*)(out + (size_t)b * D_FEAT);
  o[lane]      = r0;
  o[lane + 32] = r1;
}

extern "C" void kernel_launch(void* const* d_in, const int* in_sizes, int n_in,
                              void* d_out, int out_size, void* d_ws, size_t ws_size,
                              hipStream_t stream) {
  const int*   neigh = (const int*)d_in[0];    // [B, 32] int32
  const float* feat  = (const float*)d_in[1];  // [N, 256] f32
  float*       out   = (float*)d_out;          // [B, 256] f32

  const int Brows = in_sizes[0] / K_NEIGH;
  const int grid  = (Brows + WAVES - 1) / WAVES;
  hipLaunchKernelGGL(gnn_mean_relu_kernel, dim3(grid), dim3(WAVES * 32), 0, stream,
                     neigh, feat, out, Brows);
}